// Attention_Learngene_27161373180297
// MI455X (gfx1250) — compile-verified
//
#include <hip/hip_runtime.h>

// ---------------------------------------------------------------------------
// Problem constants (from reference)
// ---------------------------------------------------------------------------
#define B_     8
#define N_     1024
#define C_     768
#define HLG    6
#define HMID   8
#define HDESC  12
#define HD_    64
#define NH     (N_ * HD_)     // 65536
#define NN     (N_ * N_)      // 1048576
#define SCALE_ 0.125f         // 64^-0.5

typedef __attribute__((ext_vector_type(2))) float v2f;
typedef __attribute__((ext_vector_type(8))) float v8f;

__device__ __forceinline__ v8f wmma_f32(v2f a, v2f b, v8f c) {
  // D = A(16x4) * B(4x16) + C(16x16), full f32
  return __builtin_amdgcn_wmma_f32_16x16x4_f32(false, a, false, b,
                                               (short)0, c, false, false);
}

// ---------------------------------------------------------------------------
// K1: qkv = x @ qkv_w^T, scattered into (t, b, h, n, d) layout.
// Block: 256 thr = 8 waves (2 M x 4 N), tile 32x64, K-chunks of 32.
// ---------------------------------------------------------------------------
__global__ __launch_bounds__(256) void k_qkv_gemm(const float* __restrict__ x,
                                                  const float* __restrict__ w,
                                                  float* __restrict__ qkv) {
  __shared__ float As[32][36];
  __shared__ float Bs[32][68];
  const int tid  = threadIdx.x;
  const int lane = tid & 31;
  const int wid  = tid >> 5;
  const int row  = lane & 15;
  const int hi   = lane >> 4;
  const int m0w  = (wid >> 2) * 16;
  const int n0w  = (wid & 3) * 16;
  const int gm0  = blockIdx.x * 32;   // row in [0, 8192)
  const int gn0  = blockIdx.y * 64;   // col in [0, 1152)

  v8f acc = {0.f,0.f,0.f,0.f,0.f,0.f,0.f,0.f};
  for (int k0 = 0; k0 < C_; k0 += 32) {
    {
      int r = tid >> 3, c4 = (tid & 7) * 4;
      *(float4*)&As[r][c4] = *(const float4*)(x + (size_t)(gm0 + r) * C_ + k0 + c4);
    }
    {
      int j = tid >> 2, kq = (tid & 3) * 8;
      float4 v0 = *(const float4*)(w + (size_t)(gn0 + j) * C_ + k0 + kq);
      float4 v1 = *(const float4*)(w + (size_t)(gn0 + j) * C_ + k0 + kq + 4);
      Bs[kq+0][j] = v0.x; Bs[kq+1][j] = v0.y; Bs[kq+2][j] = v0.z; Bs[kq+3][j] = v0.w;
      Bs[kq+4][j] = v1.x; Bs[kq+5][j] = v1.y; Bs[kq+6][j] = v1.z; Bs[kq+7][j] = v1.w;
    }
    __syncthreads();
#pragma unroll
    for (int kk = 0; kk < 32; kk += 4) {
      v2f a, bf;
      a[0]  = As[m0w + row][kk + 2*hi];
      a[1]  = As[m0w + row][kk + 2*hi + 1];
      bf[0] = Bs[kk + 2*hi][n0w + row];
      bf[1] = Bs[kk + 2*hi + 1][n0w + row];
      acc = wmma_f32(a, bf, acc);
    }
    __syncthreads();
  }
  // 64-col tile maps to exactly one (t, h); scatter rows into (b, n)
  const int t = gn0 / 384;
  const int h = (gn0 % 384) / 64;
  const int d = n0w + row;
#pragma unroll
  for (int i = 0; i < 8; ++i) {
    int r = gm0 + m0w + i + 8*hi;
    int b = r >> 10, n = r & 1023;
    qkv[(size_t)((t * B_ + b) * HLG + h) * NH + n * HD_ + d] = acc[i];
  }
}

// ---------------------------------------------------------------------------
// K2: cross-head MLP on values: v(b,h,n,d) -> v2(b,hh,n,d)
// ---------------------------------------------------------------------------
__global__ __launch_bounds__(256) void k_vmlp(const float* __restrict__ qkv,
                                              const float* __restrict__ w1,
                                              const float* __restrict__ b1,
                                              const float* __restrict__ w2,
                                              const float* __restrict__ b2,
                                              float* __restrict__ v2out) {
  const int idx = blockIdx.x * 256 + threadIdx.x;  // [0, B*N*HD)
  const int b   = idx >> 16;
  const int nd  = idx & 65535;
  const float* vbase = qkv + (size_t)((2 * B_ + b) * HLG) * NH + nd;
  float s[HLG];
#pragma unroll
  for (int h = 0; h < HLG; ++h) s[h] = vbase[(size_t)h * NH];
  float hid[HMID];
#pragma unroll
  for (int j = 0; j < HMID; ++j) {
    float a = b1[j];
#pragma unroll
    for (int h = 0; h < HLG; ++h) a += w1[j*HLG + h] * s[h];
    hid[j] = fmaxf(a, 0.f);
  }
  float* obase = v2out + (size_t)(b * HDESC) * NH + nd;
#pragma unroll
  for (int hh = 0; hh < HDESC; ++hh) {
    float a = b2[hh];
#pragma unroll
    for (int j = 0; j < HMID; ++j) a += w2[hh*HMID + j] * hid[j];
    obase[(size_t)hh * NH] = a;
  }
}

// ---------------------------------------------------------------------------
// K3: scores s_h = q_h . k_h^T (6 heads, WMMA over K=HD=64), then 6->8->12
// cross-head MLP per (n,m) element, raw logits -> attn output region.
// One wave per block; block loops 8 m-tiles. LDS ~53 KB.
// ---------------------------------------------------------------------------
__global__ __launch_bounds__(32) void k_scores(const float* __restrict__ qkv,
                                               const float* __restrict__ w1,
                                               const float* __restrict__ b1v,
                                               const float* __restrict__ w2,
                                               const float* __restrict__ b2v,
                                               float* __restrict__ attn) {
  __shared__ float qs[HLG][16][68];
  __shared__ float ks[HLG][16][68];
  __shared__ float wgt[164]; // w1[48] b1[8] w2[96] b2[12]
  const int lane = threadIdx.x;
  const int nt = blockIdx.x;   // n-tile 0..63
  const int b  = blockIdx.y;   // batch 0..7
  const int mg = blockIdx.z;   // m-group 0..7

  for (int i = lane; i < 48; i += 32)  wgt[i]       = w1[i];
  for (int i = lane; i < 8;  i += 32)  wgt[48 + i]  = b1v[i];
  for (int i = lane; i < 96; i += 32)  wgt[56 + i]  = w2[i];
  for (int i = lane; i < 12; i += 32)  wgt[152 + i] = b2v[i];

  const float* qbase = qkv + (size_t)(b * HLG) * NH;           // t=0
  const float* kbase = qkv + (size_t)((B_ + b) * HLG) * NH;    // t=1
  for (int t4 = lane; t4 < HLG * 16 * 16; t4 += 32) {
    int h = t4 >> 8, r = (t4 >> 4) & 15, c4 = (t4 & 15) * 4;
    *(float4*)&qs[h][r][c4] =
        *(const float4*)(qbase + (size_t)h * NH + (nt*16 + r) * HD_ + c4);
  }
  const int row = lane & 15, hi = lane >> 4;

  for (int it = 0; it < 8; ++it) {
    const int mt = mg * 8 + it;
    for (int t4 = lane; t4 < HLG * 16 * 16; t4 += 32) {
      int h = t4 >> 8, r = (t4 >> 4) & 15, c4 = (t4 & 15) * 4;
      *(float4*)&ks[h][r][c4] =
          *(const float4*)(kbase + (size_t)h * NH + (mt*16 + r) * HD_ + c4);
    }
    // s_h = q_h (16xHD) . k_h^T (HDx16): B[kd][m] = k[m][kd]
    v8f acc[HLG];
#pragma unroll
    for (int h = 0; h < HLG; ++h) {
      v8f c = {0.f,0.f,0.f,0.f,0.f,0.f,0.f,0.f};
#pragma unroll
      for (int kk = 0; kk < HD_; kk += 4) {
        v2f a, bf;
        a[0]  = qs[h][row][kk + 2*hi];
        a[1]  = qs[h][row][kk + 2*hi + 1];
        bf[0] = ks[h][row][kk + 2*hi];
        bf[1] = ks[h][row][kk + 2*hi + 1];
        c = wmma_f32(a, bf, c);
      }
      acc[h] = c;
    }
    // per-element MLP; D element i sits at (n = i+8*hi, m = lane&15)
    const size_t abase = (size_t)b * HDESC * NN + (size_t)(nt * 16) * N_ + mt * 16;
#pragma unroll
    for (int i = 0; i < 8; ++i) {
      float s[HLG], hid[HMID];
#pragma unroll
      for (int h = 0; h < HLG; ++h) s[h] = acc[h][i] * SCALE_;
#pragma unroll
      for (int j = 0; j < HMID; ++j) {
        float a = wgt[48 + j];
#pragma unroll
        for (int h = 0; h < HLG; ++h) a += wgt[j*HLG + h] * s[h];
        hid[j] = fmaxf(a, 0.f);
      }
      float* abp = attn + abase + (size_t)(i + 8*hi) * N_ + row;
#pragma unroll
      for (int hh = 0; hh < HDESC; ++hh) {
        float o = wgt[152 + hh];
#pragma unroll
        for (int j = 0; j < HMID; ++j) o += wgt[56 + hh*HMID + j] * hid[j];
        abp[(size_t)hh * NN] = o;
      }
    }
  }
}

// ---------------------------------------------------------------------------
// K4: in-place row softmax over attn (8*12*1024 rows of 1024). Wave per row.
// ---------------------------------------------------------------------------
__global__ __launch_bounds__(256) void k_softmax(float* __restrict__ attn) {
  const int lane = threadIdx.x & 31, wid = threadIdx.x >> 5;
  float* p = attn + ((size_t)blockIdx.x * 8 + wid) * N_;
  float4 v[8];
  float mx = -3.4e38f;
#pragma unroll
  for (int c = 0; c < 8; ++c) {
    v[c] = *(const float4*)(p + c*128 + lane*4);
    mx = fmaxf(mx, fmaxf(fmaxf(v[c].x, v[c].y), fmaxf(v[c].z, v[c].w)));
  }
#pragma unroll
  for (int off = 16; off > 0; off >>= 1) mx = fmaxf(mx, __shfl_xor(mx, off, 32));
  float sum = 0.f;
#pragma unroll
  for (int c = 0; c < 8; ++c) {
    v[c].x = __expf(v[c].x - mx); v[c].y = __expf(v[c].y - mx);
    v[c].z = __expf(v[c].z - mx); v[c].w = __expf(v[c].w - mx);
    sum += v[c].x + v[c].y + v[c].z + v[c].w;
  }
#pragma unroll
  for (int off = 16; off > 0; off >>= 1) sum += __shfl_xor(sum, off, 32);
  const float inv = 1.0f / sum;
#pragma unroll
  for (int c = 0; c < 8; ++c) {
    v[c].x *= inv; v[c].y *= inv; v[c].z *= inv; v[c].w *= inv;
    *(float4*)(p + c*128 + lane*4) = v[c];
  }
}

// ---------------------------------------------------------------------------
// K5: out = attn @ v2 per (b,hh): (1024x1024)@(1024x64), written directly
// into hidden (B, N, 768) layout. Block tile 32x64 like K1.
// ---------------------------------------------------------------------------
__global__ __launch_bounds__(256) void k_av(const float* __restrict__ attn,
                                            const float* __restrict__ v2in,
                                            float* __restrict__ hidden) {
  __shared__ float As[32][36];
  __shared__ float Bs[32][68];
  const int tid  = threadIdx.x;
  const int lane = tid & 31, wid = tid >> 5;
  const int row = lane & 15, hi = lane >> 4;
  const int m0w = (wid >> 2) * 16, n0w = (wid & 3) * 16;
  const int gm0 = blockIdx.x * 32;   // n rows
  const int bh  = blockIdx.y;        // b*12+hh
  const float* A  = attn + (size_t)bh * NN;
  const float* Bv = v2in + (size_t)bh * NH;

  v8f acc = {0.f,0.f,0.f,0.f,0.f,0.f,0.f,0.f};
  for (int k0 = 0; k0 < N_; k0 += 32) {
    {
      int r = tid >> 3, c4 = (tid & 7) * 4;
      *(float4*)&As[r][c4] = *(const float4*)(A + (size_t)(gm0 + r) * N_ + k0 + c4);
    }
#pragma unroll
    for (int u = 0; u < 2; ++u) {
      int idx4 = tid * 2 + u;
      int k = idx4 >> 4, j4 = (idx4 & 15) * 4;
      *(float4*)&Bs[k][j4] = *(const float4*)(Bv + (size_t)(k0 + k) * HD_ + j4);
    }
    __syncthreads();
#pragma unroll
    for (int kk = 0; kk < 32; kk += 4) {
      v2f a, bf;
      a[0]  = As[m0w + row][kk + 2*hi];
      a[1]  = As[m0w + row][kk + 2*hi + 1];
      bf[0] = Bs[kk + 2*hi][n0w + row];
      bf[1] = Bs[kk + 2*hi + 1][n0w + row];
      acc = wmma_f32(a, bf, acc);
    }
    __syncthreads();
  }
  const int b = bh / HDESC, hh = bh % HDESC;
  const int col = n0w + row;  // d
#pragma unroll
  for (int i = 0; i < 8; ++i) {
    int n = gm0 + m0w + i + 8*hi;
    hidden[(size_t)(b * N_ + n) * C_ + hh * HD_ + col] = acc[i];
  }
}

// ---------------------------------------------------------------------------
// K6: out = hidden @ proj_w^T + proj_b  (8192x768 @ 768x768)
// ---------------------------------------------------------------------------
__global__ __launch_bounds__(256) void k_proj(const float* __restrict__ hidden,
                                              const float* __restrict__ w,
                                              const float* __restrict__ bias,
                                              float* __restrict__ out) {
  __shared__ float As[32][36];
  __shared__ float Bs[32][68];
  const int tid  = threadIdx.x;
  const int lane = tid & 31, wid = tid >> 5;
  const int row = lane & 15, hi = lane >> 4;
  const int m0w = (wid >> 2) * 16, n0w = (wid & 3) * 16;
  const int gm0 = blockIdx.x * 32;
  const int gn0 = blockIdx.y * 64;

  v8f acc = {0.f,0.f,0.f,0.f,0.f,0.f,0.f,0.f};
  for (int k0 = 0; k0 < C_; k0 += 32) {
    {
      int r = tid >> 3, c4 = (tid & 7) * 4;
      *(float4*)&As[r][c4] = *(const float4*)(hidden + (size_t)(gm0 + r) * C_ + k0 + c4);
    }
    {
      int j = tid >> 2, kq = (tid & 3) * 8;
      float4 v0 = *(const float4*)(w + (size_t)(gn0 + j) * C_ + k0 + kq);
      float4 v1 = *(const float4*)(w + (size_t)(gn0 + j) * C_ + k0 + kq + 4);
      Bs[kq+0][j] = v0.x; Bs[kq+1][j] = v0.y; Bs[kq+2][j] = v0.z; Bs[kq+3][j] = v0.w;
      Bs[kq+4][j] = v1.x; Bs[kq+5][j] = v1.y; Bs[kq+6][j] = v1.z; Bs[kq+7][j] = v1.w;
    }
    __syncthreads();
#pragma unroll
    for (int kk = 0; kk < 32; kk += 4) {
      v2f a, bf;
      a[0]  = As[m0w + row][kk + 2*hi];
      a[1]  = As[m0w + row][kk + 2*hi + 1];
      bf[0] = Bs[kk + 2*hi][n0w + row];
      bf[1] = Bs[kk + 2*hi + 1][n0w + row];
      acc = wmma_f32(a, bf, acc);
    }
    __syncthreads();
  }
  const int gcol = gn0 + n0w + row;
  const float pb = bias[gcol];
#pragma unroll
  for (int i = 0; i < 8; ++i) {
    int r = gm0 + m0w + i + 8*hi;
    out[(size_t)r * C_ + gcol] = acc[i] + pb;
  }
}

// ---------------------------------------------------------------------------
extern "C" void kernel_launch(void* const* d_in, const int* in_sizes, int n_in,
                              void* d_out, int out_size, void* d_ws, size_t ws_size,
                              hipStream_t stream) {
  const float* x      = (const float*)d_in[0];
  const float* qkv_w  = (const float*)d_in[1];
  const float* qk_w1  = (const float*)d_in[2];
  const float* qk_b1  = (const float*)d_in[3];
  const float* qk_w2  = (const float*)d_in[4];
  const float* qk_b2  = (const float*)d_in[5];
  const float* v_w1   = (const float*)d_in[6];
  const float* v_b1   = (const float*)d_in[7];
  const float* v_w2   = (const float*)d_in[8];
  const float* v_b2   = (const float*)d_in[9];
  const float* proj_w = (const float*)d_in[10];
  const float* proj_b = (const float*)d_in[11];

  float* out  = (float*)d_out;                       // (B,N,C) = 6291456 floats
  float* attn = out + (size_t)B_ * N_ * C_;          // (B,12,N,N) = 100663296 floats

  float* ws     = (float*)d_ws;
  float* qkv    = ws;                                // 3*B*HLG*N*HD = 9437184
  float* v2     = ws + 9437184;                      // B*HDESC*N*HD = 6291456
  float* hidden = ws + 9437184 + 6291456;            // B*N*C        = 6291456

  // 1) qkv projection (WMMA f32)
  k_qkv_gemm<<<dim3(8192 / 32, (3 * HLG * HD_) / 64), 256, 0, stream>>>(x, qkv_w, qkv);
  // 2) value cross-head MLP
  k_vmlp<<<(B_ * N_ * HD_) / 256, 256, 0, stream>>>(qkv, v_w1, v_b1, v_w2, v_b2, v2);
  // 3) scores (WMMA f32) + score cross-head MLP -> raw logits in attn region
  k_scores<<<dim3(N_ / 16, B_, 8), 32, 0, stream>>>(qkv, qk_w1, qk_b1, qk_w2, qk_b2, attn);
  // 4) in-place softmax
  k_softmax<<<(B_ * HDESC * N_) / 8, 256, 0, stream>>>(attn);
  // 5) attn @ v2 (WMMA f32) -> hidden in (B,N,C) layout
  k_av<<<dim3(N_ / 32, B_ * HDESC), 256, 0, stream>>>(attn, v2, hidden);
  // 6) output projection + bias (WMMA f32)
  k_proj<<<dim3(8192 / 32, C_ / 64), 256, 0, stream>>>(hidden, proj_w, proj_b, out);
}